// Transformer_73675868995910
// MI455X (gfx1250) — compile-verified
//
#include <hip/hip_runtime.h>
#include <math.h>

typedef __attribute__((ext_vector_type(2))) float v2f;
typedef __attribute__((ext_vector_type(8))) float v8f;

#define NQ   4096
#define NK   32768
#define DIN  1024
#define DOUT 512
#define MAXM 64          // cap on matches per query row (Poisson mean 8 -> P(>64) ~ 0)

// ---------------------------------------------------------------------------
// wave32 reductions
// ---------------------------------------------------------------------------
__device__ __forceinline__ float waveSum(float v) {
  #pragma unroll
  for (int off = 16; off > 0; off >>= 1) v += __shfl_xor(v, off, 32);
  return v;
}
__device__ __forceinline__ float waveMax(float v) {
  #pragma unroll
  for (int off = 16; off > 0; off >>= 1) v = fmaxf(v, __shfl_xor(v, off, 32));
  return v;
}

// ---------------------------------------------------------------------------
// Y = X @ W + b   (X: M x 1024, W: 1024 x 512, Y: M x 512)
// One wave computes one 16x16 output tile with V_WMMA_F32_16X16X4_F32.
// A frag (16x4 f32): lane L<16 -> row m0+L, k's {k,k+1}; lanes 16..31 -> {k+2,k+3}
// B frag (4x16 f32): mirrored, column n0+(L&15)
// C/D:  lane column n0+(L&15); vgpr r holds rows r (lanes<16) / r+8 (lanes>=16)
// ---------------------------------------------------------------------------
__global__ __launch_bounds__(256) void gemm_bias(
    const float* __restrict__ X, const float* __restrict__ W,
    const float* __restrict__ b, float* __restrict__ Y) {
  const int lane = threadIdx.x & 31;
  const int wave = threadIdx.x >> 5;
  const int m0 = blockIdx.x * 16;
  const int n0 = (blockIdx.y * 8 + wave) * 16;
  const int ll = lane & 15, lh = lane >> 4;
  const float* __restrict__ xrow = X + (size_t)(m0 + ll) * DIN;

  v8f c = {};
  for (int k = 0; k < DIN; k += 4) {
    const int ka = k + 2 * lh;
    v2f a, bb;
    a.x  = xrow[ka];
    a.y  = xrow[ka + 1];
    bb.x = W[(size_t)ka * DOUT + n0 + ll];
    bb.y = W[(size_t)(ka + 1) * DOUT + n0 + ll];
    c = __builtin_amdgcn_wmma_f32_16x16x4_f32(false, a, false, bb,
                                              (short)0, c, false, false);
  }
  const float bias = b[n0 + ll];
  #pragma unroll
  for (int r = 0; r < 8; ++r)
    Y[(size_t)(m0 + r + 8 * lh) * DOUT + n0 + ll] = c[r] + bias;
}

// ---------------------------------------------------------------------------
// Fused K/V projection: reads X2 once, shares the A fragment across two WMMAs.
// ---------------------------------------------------------------------------
__global__ __launch_bounds__(256) void gemm_bias_kv(
    const float* __restrict__ X,
    const float* __restrict__ Wk, const float* __restrict__ bk,
    const float* __restrict__ Wv, const float* __restrict__ bv,
    float* __restrict__ K, float* __restrict__ V) {
  const int lane = threadIdx.x & 31;
  const int wave = threadIdx.x >> 5;
  const int m0 = blockIdx.x * 16;
  const int n0 = (blockIdx.y * 8 + wave) * 16;
  const int ll = lane & 15, lh = lane >> 4;
  const float* __restrict__ xrow = X + (size_t)(m0 + ll) * DIN;

  v8f ck = {}, cv = {};
  for (int k = 0; k < DIN; k += 4) {
    const int ka = k + 2 * lh;
    v2f a, bfk, bfv;
    a.x   = xrow[ka];
    a.y   = xrow[ka + 1];
    bfk.x = Wk[(size_t)ka * DOUT + n0 + ll];
    bfk.y = Wk[(size_t)(ka + 1) * DOUT + n0 + ll];
    bfv.x = Wv[(size_t)ka * DOUT + n0 + ll];
    bfv.y = Wv[(size_t)(ka + 1) * DOUT + n0 + ll];
    ck = __builtin_amdgcn_wmma_f32_16x16x4_f32(false, a, false, bfk,
                                               (short)0, ck, false, false);
    cv = __builtin_amdgcn_wmma_f32_16x16x4_f32(false, a, false, bfv,
                                               (short)0, cv, false, false);
  }
  const float biask = bk[n0 + ll];
  const float biasv = bv[n0 + ll];
  #pragma unroll
  for (int r = 0; r < 8; ++r) {
    const size_t idx = (size_t)(m0 + r + 8 * lh) * DOUT + n0 + ll;
    K[idx] = ck[r] + biask;
    V[idx] = cv[r] + biasv;
  }
}

// ---------------------------------------------------------------------------
// s[j] = (Q[row_map[j]] . K[j]) / sqrt(512)   — one wave per key j
// ---------------------------------------------------------------------------
__global__ __launch_bounds__(256) void score_kernel(
    const float* __restrict__ Q, const float* __restrict__ K,
    const int* __restrict__ row_map, float* __restrict__ s) {
  const int lane = threadIdx.x & 31;
  const int j = blockIdx.x * 8 + (threadIdx.x >> 5);
  const int r = row_map[j];
  const float* __restrict__ q = Q + (size_t)r * DOUT;
  const float* __restrict__ kk = K + (size_t)j * DOUT;
  float acc = 0.f;
  #pragma unroll
  for (int t = 0; t < DOUT / 32; ++t)
    acc = fmaf(q[lane + 32 * t], kk[lane + 32 * t], acc);
  acc = waveSum(acc);
  if (lane == 0) s[j] = acc * 0.04419417382415922f;  // 1/sqrt(512)
}

// ---------------------------------------------------------------------------
// One wave per query row i: find {j : row_map[j]==i} (deterministic j order),
// softmax over its scores (exp(-1000-m) underflows to 0 in fp32, so masked
// entries contribute nothing), then C_i = sum w_j * V[j]. Rows with no
// matches get zeros (matches the counts>0 select in the reference).
// ---------------------------------------------------------------------------
__global__ __launch_bounds__(256) void attend_kernel(
    const float* __restrict__ s, const float* __restrict__ V,
    const int* __restrict__ row_map, float* __restrict__ out) {
  __shared__ int midx[8][MAXM];
  const int lane = threadIdx.x & 31;
  const int wave = threadIdx.x >> 5;
  const int i = blockIdx.x * 8 + wave;

  int cnt = 0;
  for (int jb = lane; jb < NK; jb += 32) {       // uniform trip count
    const bool hit = (row_map[jb] == i);
    const unsigned mask = (unsigned)__ballot(hit);
    if (hit) {
      const int pos = cnt + __popc(mask & ((1u << lane) - 1u));
      if (pos < MAXM) midx[wave][pos] = jb;
    }
    cnt += __popc(mask);
  }
  if (cnt > MAXM) cnt = MAXM;

  if (cnt == 0) {
    #pragma unroll
    for (int t = 0; t < DOUT / 32; ++t)
      out[(size_t)i * DOUT + lane + 32 * t] = 0.f;
    return;
  }

  float m = -3.0e38f;
  for (int l = lane; l < cnt; l += 32) m = fmaxf(m, s[midx[wave][l]]);
  m = waveMax(m);

  float Z = 0.f;
  for (int l = lane; l < cnt; l += 32) Z += __expf(s[midx[wave][l]] - m);
  Z = waveSum(Z);
  const float invZ = 1.0f / Z;

  float acc[DOUT / 32];
  #pragma unroll
  for (int t = 0; t < DOUT / 32; ++t) acc[t] = 0.f;

  for (int l = 0; l < cnt; ++l) {
    const int jl = midx[wave][l];
    const float w = __expf(s[jl] - m) * invZ;    // uniform across the wave
    const float* __restrict__ v = V + (size_t)jl * DOUT;
    #pragma unroll
    for (int t = 0; t < DOUT / 32; ++t)
      acc[t] = fmaf(w, v[lane + 32 * t], acc[t]);
  }
  #pragma unroll
  for (int t = 0; t < DOUT / 32; ++t)
    out[(size_t)i * DOUT + lane + 32 * t] = acc[t];
}

// ---------------------------------------------------------------------------
extern "C" void kernel_launch(void* const* d_in, const int* in_sizes, int n_in,
                              void* d_out, int out_size, void* d_ws, size_t ws_size,
                              hipStream_t stream) {
  (void)in_sizes; (void)n_in; (void)out_size; (void)ws_size;
  const float* X1      = (const float*)d_in[0];
  const float* X2      = (const float*)d_in[1];
  const float* Wq      = (const float*)d_in[2];
  const float* bq      = (const float*)d_in[3];
  const float* Wk      = (const float*)d_in[4];
  const float* bk      = (const float*)d_in[5];
  const float* Wv      = (const float*)d_in[6];
  const float* bv      = (const float*)d_in[7];
  const int*   row_map = (const int*)d_in[8];
  float* out = (float*)d_out;

  // workspace layout: Q (8 MB) | K (64 MB) | V (64 MB) | s (128 KB)
  char* ws = (char*)d_ws;
  float* Q = (float*)ws;
  float* K = (float*)(ws + (size_t)NQ * DOUT * sizeof(float));
  float* V = (float*)(ws + (size_t)(NQ + NK) * DOUT * sizeof(float));
  float* s = (float*)(ws + (size_t)(NQ + 2 * (size_t)NK) * DOUT * sizeof(float));

  const dim3 blk(256);
  gemm_bias   <<<dim3(NQ / 16, DOUT / 16 / 8), blk, 0, stream>>>(X1, Wq, bq, Q);
  gemm_bias_kv<<<dim3(NK / 16, DOUT / 16 / 8), blk, 0, stream>>>(X2, Wk, bk, Wv, bv, K, V);
  score_kernel<<<dim3(NK / 8), blk, 0, stream>>>(Q, K, row_map, s);
  attend_kernel<<<dim3(NQ / 8), blk, 0, stream>>>(s, V, row_map, out);
}